// MCGNN_42941083026054
// MI455X (gfx1250) — compile-verified
//
#include <hip/hip_runtime.h>
#include <cstdint>
#include <cstddef>

// ---------------------------------------------------------------------------
// MCGNN pipeline for MI455X (gfx1250, wave32).
// Dense matmuls use V_WMMA_F32_16X16X4_F32 (fp32-preserving matrix path).
// K=128/N=128 GEMMs stage the 64KB weight matrix in LDS via
// GLOBAL_LOAD_ASYNC_TO_LDS_B128 (ASYNCcnt) and feed WMMA from ds_load.
// Edge (GAT) softmax-aggregation uses hardware fp32 atomics + ordered-uint max.
// ---------------------------------------------------------------------------

#define DD 128

typedef __attribute__((ext_vector_type(2))) float v2f;
typedef __attribute__((ext_vector_type(8))) float v8f;

// ===========================================================================
// Generic fp32 WMMA GEMM (global-B path): C[M,N] = act((accum?C:0)+A@B+bias)
// Used only for the N=512 cosine-similarity GEMM.
// ===========================================================================
__global__ void gemm_wmma_f32(const float* __restrict__ A,
                              const float* __restrict__ Bm,
                              const float* __restrict__ bias,
                              const int*   __restrict__ gather,
                              float* __restrict__ C,
                              int M, int N, int K, int lda,
                              int accum, int act)
{
    int wave  = (int)((blockIdx.x * blockDim.x + threadIdx.x) >> 5);
    int lane  = (int)(threadIdx.x & 31);
    int tilesN = N >> 4;
    int tM = (wave / tilesN) << 4;
    int tN = (wave % tilesN) << 4;
    if (tM >= M) return;                       // wave-uniform exit

    int  mrow = tM + (lane & 15);
    int  rr   = mrow < M ? mrow : (M - 1);     // clamp; zero via mask (no branches)
    float msk = mrow < M ? 1.f : 0.f;
    long r    = gather ? (long)gather[rr] : (long)rr;
    int  kh   = (lane >> 4) << 1;              // 0 or 2
    const float* ap = A + r * (long)lda + kh;
    int ncol = tN + (lane & 15);

    v8f acc = {0.f,0.f,0.f,0.f,0.f,0.f,0.f,0.f};
    for (int k = 0; k < K; k += 4) {
        v2f a; a[0] = ap[k] * msk; a[1] = ap[k + 1] * msk;
        const float* bp = Bm + (long)(k + kh) * N + ncol;
        v2f b; b[0] = bp[0]; b[1] = bp[N];
        acc = __builtin_amdgcn_wmma_f32_16x16x4_f32(
                  false, a, false, b, (short)0, acc, false, false);
    }
    int r0 = tM + ((lane >> 4) << 3);
#pragma unroll
    for (int i = 0; i < 8; ++i) {
        int row = r0 + i;
        if (row < M) {
            long off = (long)row * N + ncol;
            float v = acc[i];
            if (accum) v += C[off];
            if (bias)  v += bias[ncol];
            if (act == 1)      v = 1.f / (1.f + __expf(-v));
            else if (act == 2) v = tanhf(v);
            C[off] = v;
        }
    }
}

// ===========================================================================
// LDS-staged WMMA GEMM, fixed N=K=128.
// Block = 256 threads (8 waves) covers 128 rows; B (64KB) is async-copied to
// LDS once per block; wave w computes rows [tM128+16w, +16) x all 128 cols.
// A dynamically-dead store to Bs keeps LLVM's LDS-global optimization from
// folding the asm-written LDS reads to poison (asm is the only other writer).
// ===========================================================================
__global__ __launch_bounds__(256) void gemm_wmma_lds128(
    const float* __restrict__ A, const float* __restrict__ Bm,
    const float* __restrict__ bias, const int* __restrict__ gather,
    float* __restrict__ C, int M, int lda, int accum, int act)
{
    __shared__ float Bs[DD * DD];              // 64 KB of the 320 KB WGP LDS
    int tid = (int)threadIdx.x;

    // Never taken at runtime (all call sites pass M > 0); the compiler cannot
    // prove that, so Bs has a visible store and its loads stay materialized.
    if (M == (-2147483647 - 1)) Bs[tid] = (float)lda;

    // ---- async stage B -> LDS: 16 rounds x 256 threads x 16B = 64KB ----
    // Bs is the only LDS object => wave LDS base offset 0.
#pragma unroll
    for (int rnd = 0; rnd < 16; ++rnd) {
        unsigned lds_off = (unsigned)((rnd * 256 + tid) * 16);
        unsigned long long ga =
            (unsigned long long)(Bm + (size_t)(rnd * 256 + tid) * 4);
        asm volatile("global_load_async_to_lds_b128 %0, %1, off"
                     :: "v"(lds_off), "v"(ga) : "memory");
    }
    asm volatile("s_wait_asynccnt 0x0" ::: "memory");
    __syncthreads();

    int wave = tid >> 5, lane = tid & 31;
    int tM = (int)blockIdx.x * 128 + wave * 16;
    if (tM >= M) return;                       // wave-uniform (after barrier)

    int  mrow = tM + (lane & 15);
    int  rr   = mrow < M ? mrow : (M - 1);
    float msk = mrow < M ? 1.f : 0.f;
    long r    = gather ? (long)gather[rr] : (long)rr;
    int  kh   = (lane >> 4) << 1;              // 0 or 2
    const float* ap = A + r * (long)lda + kh;
    int nc = lane & 15;

    v8f acc[8];
#pragma unroll
    for (int j = 0; j < 8; ++j) acc[j] = (v8f){0.f,0.f,0.f,0.f,0.f,0.f,0.f,0.f};

    for (int k = 0; k < DD; k += 4) {
        v2f a; a[0] = ap[k] * msk; a[1] = ap[k + 1] * msk;
        int bidx = (k + kh) * DD + nc;         // direct LDS indexing -> ds_load
#pragma unroll
        for (int j = 0; j < 8; ++j) {
            v2f b; b[0] = Bs[bidx + j * 16]; b[1] = Bs[bidx + j * 16 + DD];
            acc[j] = __builtin_amdgcn_wmma_f32_16x16x4_f32(
                         false, a, false, b, (short)0, acc[j], false, false);
        }
    }

    int r0 = tM + ((lane >> 4) << 3);
#pragma unroll
    for (int j = 0; j < 8; ++j) {
        int col = j * 16 + nc;
#pragma unroll
        for (int i = 0; i < 8; ++i) {
            int row = r0 + i;
            if (row < M) {
                long off = (long)row * DD + col;
                float v = acc[j][i];
                if (accum) v += C[off];
                if (bias)  v += bias[col];
                if (act == 1)      v = 1.f / (1.f + __expf(-v));
                else if (act == 2) v = tanhf(v);
                C[off] = v;
            }
        }
    }
}

// ===========================================================================
// Elementwise / gather / GAT helper kernels
// ===========================================================================
__global__ void k_fill_f32(float* p, float v, long n) {
    long i = (long)blockIdx.x * blockDim.x + threadIdx.x;
    if (i < n) p[i] = v;
}
__global__ void k_fill_u32(unsigned* p, unsigned v, long n) {
    long i = (long)blockIdx.x * blockDim.x + threadIdx.x;
    if (i < n) p[i] = v;
}
__global__ void k_gather_rows(float* dst, const float* src, const int* idx, long n) {
    long i = (long)blockIdx.x * blockDim.x + threadIdx.x;
    if (i >= n) return;
    long r = i >> 7; int d = (int)(i & 127);
    dst[i] = src[(long)idx[r] * DD + d];
}
__global__ void k_init_node(const float* h, const float* b1, const float* b2,
                            float* ni, long n) {
    long i = (long)blockIdx.x * blockDim.x + threadIdx.x;
    if (i >= n) return;
    int d = (int)(i & 127);
    ni[i] = 2.f * h[i] + b1[d] + b2[d];
}
__global__ void k_rowdot(const float* Z, const float* v, float* s, int n) {
    int wid  = (int)((blockIdx.x * blockDim.x + threadIdx.x) >> 5);
    int lane = (int)(threadIdx.x & 31);
    if (wid >= n) return;
    const float* row = Z + (long)wid * DD;
    float sum = 0.f;
    for (int i = lane; i < DD; i += 32) sum += row[i] * v[i];
    for (int o = 16; o > 0; o >>= 1) sum += __shfl_down(sum, o, 32);
    if (lane == 0) s[wid] = sum;
}
__device__ __forceinline__ unsigned f2ord(float f) {
    unsigned u = __float_as_uint(f);
    return (u & 0x80000000u) ? ~u : (u | 0x80000000u);
}
__device__ __forceinline__ float ord2f(unsigned u) {
    unsigned b = (u & 0x80000000u) ? (u & 0x7FFFFFFFu) : ~u;
    return __uint_as_float(b);
}
__global__ void k_edge1(const float* sl, const float* sr, const int* src,
                        const int* dst, float* e, unsigned* mord, long E) {
    long i = (long)blockIdx.x * blockDim.x + threadIdx.x;
    if (i >= E) return;
    float x = sl[src[i]] + sr[dst[i]];
    x = x > 0.f ? x : 0.2f * x;
    e[i] = x;
    atomicMax(mord + dst[i], f2ord(x));
}
__global__ void k_edge2(float* e, const int* dst, const unsigned* mord,
                        float* den, long E) {
    long i = (long)blockIdx.x * blockDim.x + threadIdx.x;
    if (i >= E) return;
    float ee = __expf(e[i] - ord2f(mord[dst[i]]));
    e[i] = ee;
    unsafeAtomicAdd(den + dst[i], ee);
}
__global__ void k_edge3(const float* e, const float* den, const int* src,
                        const int* dst, const float* Z, float* out, long tot) {
    long i = (long)blockIdx.x * blockDim.x + threadIdx.x;
    if (i >= tot) return;
    long edge = i >> 7; int d = (int)(i & 127);
    int sN = src[edge], dN = dst[edge];
    float alpha = e[edge] / den[dN];
    unsafeAtomicAdd(out + (long)dN * DD + d, alpha * Z[(long)sN * DD + d]);
}
__global__ void k_gate(const float* g, const float* h0, const float* h,
                       float* f, long n) {
    long i = (long)blockIdx.x * blockDim.x + threadIdx.x;
    if (i >= n) return;
    float gg = g[i];
    f[i] = gg * h0[i] + (1.f - gg) * h[i];
}
__global__ void k_hidden(float* hid, const float* item_emb, const int* seq,
                         const float* feat, const int* alias, const int* mask,
                         const float* xs, long n) {
    long i = (long)blockIdx.x * blockDim.x + threadIdx.x;
    if (i >= n) return;
    long bt = i >> 7; int d = (int)(i & 127);
    float m = (float)mask[bt];
    hid[i] = item_emb[(long)seq[bt] * DD + d] * xs[0]
           + feat[(long)alias[bt] * DD + d] * m;
}
__global__ void k_attn_hsv(const float* hidden, const float* q, const int* mask,
                           const float* g, const float* bv, float* out, int T) {
    int b = blockIdx.x, d = threadIdx.x;       // 128 threads
    __shared__ float red[DD];
    __shared__ float hs[64];
    const float* hb = hidden + (long)b * T * DD;
    float qd = q[d];
    for (int t = 0; t < T; ++t) {
        red[d] = hb[t * DD + d] * qd;
        __syncthreads();
        for (int s = 64; s > 0; s >>= 1) { if (d < s) red[d] += red[d + s]; __syncthreads(); }
        if (d == 0) hs[t] = red[0];
        __syncthreads();
    }
    float mx = -1e30f;
    for (int t = 0; t < T; ++t) mx = fmaxf(mx, hs[t]);
    float sum = 0.f;
    for (int t = 0; t < T; ++t) sum += __expf(hs[t] - mx);
    float acc = 0.f;
    for (int t = 0; t < T; ++t) {
        float beta = __expf(hs[t] - mx) / sum * (float)mask[b * T + t];
        acc += beta * hb[t * DD + d];
    }
    red[d] = acc; __syncthreads();
    for (int s = 64; s > 0; s >>= 1) { if (d < s) red[d] += red[d + s]; __syncthreads(); }
    float mu = red[0] / (float)DD; __syncthreads();
    float c = acc - mu;
    red[d] = c * c; __syncthreads();
    for (int s = 64; s > 0; s >>= 1) { if (d < s) red[d] += red[d + s]; __syncthreads(); }
    float var = red[0] / (float)DD;
    out[(long)b * DD + d] = c * rsqrtf(var + 1e-8f) * g[d] + bv[d];
}
__global__ void k_bcastP(float* nh, const float* P, long n, int T) {
    long i = (long)blockIdx.x * blockDim.x + threadIdx.x;
    if (i >= n) return;
    long bt = i >> 7; int t = (int)(bt % T); int d = (int)(i & 127);
    nh[i] = P[t * DD + d];
}
__global__ void k_glusig(float* nhg, const float* bv, const float* hsvg,
                         long n, int T) {
    long i = (long)blockIdx.x * blockDim.x + threadIdx.x;
    if (i >= n) return;
    long bt = i >> 7; long b = bt / T; int d = (int)(i & 127);
    float v = nhg[i] + bv[d] + hsvg[b * DD + d];
    nhg[i] = 1.f / (1.f + __expf(-v));
}
__global__ void k_betasel(const float* hidden, const float* nhg, const float* w2,
                          const int* mask, const float* feat, const int* last_idx,
                          const float* ys, const float* g, const float* bv,
                          float* h_all, int T) {
    int b = blockIdx.x, d = threadIdx.x;       // 128 threads
    __shared__ float red[DD];
    __shared__ float betas[64];
    const float* hb = hidden + (long)b * T * DD;
    const float* nb = nhg    + (long)b * T * DD;
    float w2d = w2[d];
    for (int t = 0; t < T; ++t) {
        red[d] = nb[t * DD + d] * w2d;
        __syncthreads();
        for (int s = 64; s > 0; s >>= 1) { if (d < s) red[d] += red[d + s]; __syncthreads(); }
        if (d == 0) betas[t] = red[0] * (float)mask[b * T + t];
        __syncthreads();
    }
    float sel = 0.f;
    for (int t = 0; t < T; ++t) sel += betas[t] * hb[t * DD + d];
    red[d] = sel; __syncthreads();
    for (int s = 64; s > 0; s >>= 1) { if (d < s) red[d] += red[d + s]; __syncthreads(); }
    float mu = red[0] / (float)DD; __syncthreads();
    float c = sel - mu;
    red[d] = c * c; __syncthreads();
    for (int s = 64; s > 0; s >>= 1) { if (d < s) red[d] += red[d + s]; __syncthreads(); }
    float var = red[0] / (float)DD;
    float hl = c * rsqrtf(var + 1e-8f) * g[d] + bv[d];
    h_all[(long)b * DD + d] = hl + feat[(long)last_idx[b] * DD + d] * ys[0];
}
__global__ void k_transpose(float* ht, const float* h, int Bn) {
    long i = (long)blockIdx.x * blockDim.x + threadIdx.x;
    if (i >= (long)Bn * DD) return;
    long b = i >> 7; int d = (int)(i & 127);
    ht[(long)d * Bn + b] = h[i];
}
__global__ void k_rownorm(const float* h, float* nrm, int Bn) {
    int b = blockIdx.x, lane = threadIdx.x;    // 32 threads
    const float* r = h + (long)b * DD;
    float s = 0.f;
    for (int i = lane; i < DD; i += 32) { float x = r[i]; s += x * x + 1e-6f; }
    for (int o = 16; o > 0; o >>= 1) s += __shfl_down(s, o, 32);
    if (lane == 0) nrm[b] = sqrtf(s);
}
__global__ void k_cosdiv(float* c, const float* nrm, int Bn) {
    long i = (long)blockIdx.x * blockDim.x + threadIdx.x;
    if (i >= (long)Bn * Bn) return;
    long r = i / Bn, col = i % Bn;
    c[i] = c[i] / (nrm[r] * nrm[col]);
}

// ===========================================================================
// Host-side orchestration
// ===========================================================================
static inline int ewb(long n) { return (int)((n + 255) / 256); }

// generic path (used for N=512 cosine GEMM)
static void gemm(hipStream_t st, const float* A, const float* Bm, const float* bias,
                 const int* gather, float* C, int M, int N, int K, int lda,
                 int accum, int act) {
    long tiles = (long)((M + 15) / 16) * (N / 16);
    int blocks = (int)((tiles + 7) / 8);
    gemm_wmma_f32<<<blocks, 256, 0, st>>>(A, Bm, bias, gather, C, M, N, K, lda, accum, act);
}
// LDS-staged path, N=K=128
static void gemm128(hipStream_t st, const float* A, const float* Bm, const float* bias,
                    const int* gather, float* C, int M, int lda, int accum, int act) {
    int blocks = (M + 127) / 128;
    gemm_wmma_lds128<<<blocks, 256, 0, st>>>(A, Bm, bias, gather, C, M, lda, accum, act);
}

static void run_gat(hipStream_t st, const float* Zsrc, const float* sl, const float* sr,
                    const int* src, const int* dst, long E, long ndst,
                    unsigned* mord, float* den, float* eb, float* out) {
    k_fill_u32<<<ewb(ndst), 256, 0, st>>>(mord, 0u, ndst);
    k_fill_f32<<<ewb(ndst), 256, 0, st>>>(den, 0.f, ndst);
    k_edge1<<<ewb(E), 256, 0, st>>>(sl, sr, src, dst, eb, mord, E);
    k_edge2<<<ewb(E), 256, 0, st>>>(eb, dst, mord, den, E);
    long tot = E * DD;
    k_edge3<<<ewb(tot), 256, 0, st>>>(eb, den, src, dst, Zsrc, out, tot);
}

extern "C" void kernel_launch(void* const* d_in, const int* in_sizes, int n_in,
                              void* d_out, int out_size, void* d_ws, size_t ws_size,
                              hipStream_t stream) {
    (void)in_sizes; (void)n_in; (void)out_size; (void)ws_size;
    const long NI = 100000, NC = 10000, B = 512, T = 50, L = 2;
    const long EII = 800000, ECC = 100000, ECI = 400000, EIC = 400000;

    const float* item_emb = (const float*)d_in[0];
    const float* cate_emb = (const float*)d_in[1];
    const float* pos_emb  = (const float*)d_in[2];
    const float* W_pos    = (const float*)d_in[3];
    const float* gat_W    = (const float*)d_in[4];
    const float* gat_al   = (const float*)d_in[5];
    const float* gat_ar   = (const float*)d_in[6];
    const float* gat_b    = (const float*)d_in[7];
    const float* Wg1_w    = (const float*)d_in[8];
    const float* Wg1_b    = (const float*)d_in[9];
    const float* q        = (const float*)d_in[12];
    const float* lin_w    = (const float*)d_in[13];
    const float* lin_b    = (const float*)d_in[14];
    const float* w_1      = (const float*)d_in[15];
    const float* w_2      = (const float*)d_in[16];
    const float* glu1_w   = (const float*)d_in[17];
    const float* glu1_b   = (const float*)d_in[18];
    const float* glu2_w   = (const float*)d_in[19];
    const float* ln1_g    = (const float*)d_in[20];
    const float* ln1_b    = (const float*)d_in[21];
    const float* ln2_g    = (const float*)d_in[22];
    const float* ln2_b    = (const float*)d_in[23];
    const float* x_s      = (const float*)d_in[24];
    const float* y_s      = (const float*)d_in[25];
    const int* iid    = (const int*)d_in[26];
    const int* icate  = (const int*)d_in[27];
    const int* cid    = (const int*)d_in[28];
    const int* src_ii = (const int*)d_in[29];
    const int* dst_ii = (const int*)d_in[30];
    const int* src_cc = (const int*)d_in[31];
    const int* dst_cc = (const int*)d_in[32];
    const int* src_ci = (const int*)d_in[33];
    const int* dst_ci = (const int*)d_in[34];
    const int* src_ic = (const int*)d_in[35];
    const int* dst_ic = (const int*)d_in[36];
    const int* alias  = (const int*)d_in[37];
    const int* seqit  = (const int*)d_in[38];
    const int* maski  = (const int*)d_in[39];
    const int* lasti  = (const int*)d_in[40];

    auto W  = [&](int l, int h) { return gat_W  + ((long)(l * 4 + h)) * DD * DD; };
    auto AL = [&](int l, int h) { return gat_al + ((long)(l * 4 + h)) * DD; };
    auto AR = [&](int l, int h) { return gat_ar + ((long)(l * 4 + h)) * DD; };
    auto BI = [&](int l, int h) { return gat_b  + ((long)(l * 4 + h)) * DD; };

    char* wsb = (char*)d_ws; size_t off = 0;
    auto alc = [&](size_t bytes) -> void* {
        void* p = wsb + off; off = (off + bytes + 255) & ~((size_t)255); return p;
    };
    float* HI0 = (float*)alc(NI * DD * 4);
    float* H1  = (float*)alc(NI * DD * 4);
    float* H2  = (float*)alc(NI * DD * 4);
    float* ZB  = (float*)alc(NI * DD * 4);
    float* HC0 = (float*)alc(NC * DD * 4);
    float* C1  = (float*)alc(NC * DD * 4);
    float* C2  = (float*)alc(NC * DD * 4);
    float* ZS  = (float*)alc(NC * DD * 4);
    float* SLB = (float*)alc(NI * 4);
    float* SRB = (float*)alc(NI * 4);
    float* SLS = (float*)alc(NC * 4);
    float* SRS = (float*)alc(NC * 4);
    float* EB  = (float*)alc(EII * 4);
    unsigned* MORD = (unsigned*)alc(NI * 4);
    float* DEN  = (float*)alc(NI * 4);
    float* HID  = (float*)alc(B * T * DD * 4);
    float* NH   = (float*)alc(B * T * DD * 4);
    float* NHG  = (float*)alc(B * T * DD * 4);
    float* HSVL = (float*)alc(B * DD * 4);
    float* HSV2 = (float*)alc(B * DD * 4);
    float* HSVG = (float*)alc(B * DD * 4);
    float* PB   = (float*)alc(T * DD * 4);
    float* NORM = (float*)alc(B * 4);
    float* HAT  = (float*)alc(DD * B * 4);

    hipStream_t st = stream;

    // ---- hi0 = [item_emb[iid] | cate_emb[icate]] @ W_pos  (split-K) ----
    gemm128(st, item_emb, W_pos,           nullptr, iid,   HI0, (int)NI, DD, 0, 0);
    gemm128(st, cate_emb, W_pos + DD * DD, nullptr, icate, HI0, (int)NI, DD, 1, 0);
    k_gather_rows<<<ewb(NC * DD), 256, 0, st>>>(HC0, cate_emb, cid, NC * DD);

    // ---- GAT layers ----
    const float* hi_cur = HI0; const float* hc_cur = HC0;
    for (int l = 0; l < (int)L; ++l) {
        float* hi_nxt = (l == 0) ? H1 : H2;
        float* hc_nxt = (l == 0) ? C1 : C2;
        k_init_node<<<ewb(NI * DD), 256, 0, st>>>(hi_cur, BI(l, 0), BI(l, 2), hi_nxt, NI * DD);
        k_init_node<<<ewb(NC * DD), 256, 0, st>>>(hc_cur, BI(l, 1), BI(l, 3), hc_nxt, NC * DD);

        // item->item
        gemm128(st, hi_cur, W(l, 0), nullptr, nullptr, ZB, (int)NI, DD, 0, 0);
        k_rowdot<<<ewb(NI * 32), 256, 0, st>>>(ZB, AL(l, 0), SLB, (int)NI);
        k_rowdot<<<ewb(NI * 32), 256, 0, st>>>(ZB, AR(l, 0), SRB, (int)NI);
        run_gat(st, ZB, SLB, SRB, src_ii, dst_ii, EII, NI, MORD, DEN, EB, hi_nxt);

        // cate->item
        gemm128(st, hi_cur, W(l, 2), nullptr, nullptr, ZB, (int)NI, DD, 0, 0);
        k_rowdot<<<ewb(NI * 32), 256, 0, st>>>(ZB, AR(l, 2), SRB, (int)NI);
        gemm128(st, hc_cur, W(l, 2), nullptr, nullptr, ZS, (int)NC, DD, 0, 0);
        k_rowdot<<<ewb(NC * 32), 256, 0, st>>>(ZS, AL(l, 2), SLS, (int)NC);
        run_gat(st, ZS, SLS, SRB, src_ci, dst_ci, ECI, NI, MORD, DEN, EB, hi_nxt);

        // cate->cate
        gemm128(st, hc_cur, W(l, 1), nullptr, nullptr, ZS, (int)NC, DD, 0, 0);
        k_rowdot<<<ewb(NC * 32), 256, 0, st>>>(ZS, AL(l, 1), SLS, (int)NC);
        k_rowdot<<<ewb(NC * 32), 256, 0, st>>>(ZS, AR(l, 1), SRS, (int)NC);
        run_gat(st, ZS, SLS, SRS, src_cc, dst_cc, ECC, NC, MORD, DEN, EB, hc_nxt);

        // item->cate
        gemm128(st, hc_cur, W(l, 3), nullptr, nullptr, ZS, (int)NC, DD, 0, 0);
        k_rowdot<<<ewb(NC * 32), 256, 0, st>>>(ZS, AR(l, 3), SRS, (int)NC);
        gemm128(st, hi_cur, W(l, 3), nullptr, nullptr, ZB, (int)NI, DD, 0, 0);
        k_rowdot<<<ewb(NI * 32), 256, 0, st>>>(ZB, AL(l, 3), SLB, (int)NI);
        run_gat(st, ZB, SLB, SRS, src_ic, dst_ic, EIC, NC, MORD, DEN, EB, hc_nxt);

        hi_cur = hi_nxt; hc_cur = hc_nxt;
    }

    // ---- item gating (feat_c is dead code in the reference; skipped) ----
    gemm128(st, HI0,    Wg1_w,           nullptr, nullptr, ZB, (int)NI, DD, 0, 0);
    gemm128(st, hi_cur, Wg1_w + DD * DD, Wg1_b,   nullptr, ZB, (int)NI, DD, 1, 1);
    float* FEAT = H1;
    k_gate<<<ewb(NI * DD), 256, 0, st>>>(ZB, HI0, hi_cur, FEAT, NI * DD);

    // ---- sequence head ----
    k_hidden<<<ewb(B * T * DD), 256, 0, st>>>(HID, item_emb, seqit, FEAT, alias,
                                              maski, x_s, B * T * DD);
    k_attn_hsv<<<(int)B, DD, 0, st>>>(HID, q, maski, ln1_g, ln1_b, HSVL, (int)T);

    gemm128(st, HSVL, lin_w,           nullptr, nullptr, HSV2, (int)B, DD, 0, 0);
    gemm128(st, HID,  lin_w + DD * DD, lin_b,   nullptr, HSV2, (int)B, (int)(T * DD), 1, 0);

    gemm128(st, pos_emb, w_1, nullptr, nullptr, PB, (int)T, DD, 0, 0);
    k_bcastP<<<ewb(B * T * DD), 256, 0, st>>>(NH, PB, B * T * DD, (int)T);
    gemm128(st, HID, w_1 + DD * DD, nullptr, nullptr, NH, (int)(B * T), DD, 1, 2);

    gemm128(st, HSV2, glu2_w, nullptr, nullptr, HSVG, (int)B, DD, 0, 0);
    gemm128(st, NH,   glu1_w, nullptr, nullptr, NHG, (int)(B * T), DD, 0, 0);
    k_glusig<<<ewb(B * T * DD), 256, 0, st>>>(NHG, glu1_b, HSVG, B * T * DD, (int)T);

    float* outF = (float*)d_out;
    k_betasel<<<(int)B, DD, 0, st>>>(HID, NHG, w_2, maski, FEAT, lasti, y_s,
                                     ln2_g, ln2_b, outF, (int)T);

    // ---- cosine similarity (generic WMMA path, N=512) ----
    k_transpose<<<ewb(B * DD), 256, 0, st>>>(HAT, outF, (int)B);
    k_rownorm<<<(int)B, 32, 0, st>>>(outF, NORM, (int)B);
    float* cosO = outF + B * DD;
    gemm(st, outF, HAT, nullptr, nullptr, cosO, (int)B, (int)B, DD, DD, 0, 0);
    k_cosdiv<<<ewb(B * B), 256, 0, st>>>(cosO, NORM, (int)B);
}